// AtomUpdateBlock_33200097198200
// MI455X (gfx1250) — compile-verified
//
#include <hip/hip_runtime.h>

typedef float v2f __attribute__((ext_vector_type(2)));
typedef float v4f __attribute__((ext_vector_type(4)));
typedef float v8f __attribute__((ext_vector_type(8)));

#define N_ATOMS 100000
#define N_EDGES 800000
#define D_ATOM  256
#define D_EDGE  256
#define D_X     512          // 2*D_EDGE (accumulator row width)
#define K_TOT   768          // D_X + D_ATOM
#define M_TILES (N_ATOMS/16) // 6250
#define KC      64           // K-chunk staged in LDS (64x256 f32 = 64KB)

// ---------------- zero the scatter accumulator (128-bit stores) ----------------
__global__ void zero_ws_kernel(v4f* __restrict__ X4, long n4) {
    long i = (long)blockIdx.x * blockDim.x + threadIdx.x;
    if (i < n4) X4[i] = (v4f){0.f, 0.f, 0.f, 0.f};
}

// ---------------- edge scatter: X[id1]+=m1, X[id3]-=m1, X[id2+off]+=m2, X[id4+off]-=m2
// Edge data is streamed exactly once -> non-temporal loads so the 205MB
// accumulator stays resident in the 192MB L2 for the atomics.
__global__ void edge_scatter_kernel(const v4f* __restrict__ m1,
                                    const v4f* __restrict__ m2,
                                    const int* __restrict__ id1,
                                    const int* __restrict__ id2,
                                    const int* __restrict__ id3,
                                    const int* __restrict__ id4,
                                    float* __restrict__ X) {
    long t = (long)blockIdx.x * blockDim.x + threadIdx.x;
    const long total = (long)N_EDGES * 64;
    if (t >= total) return;
    int e = (int)(t >> 6);
    int q = (int)(t & 63);
    int c = q * 4;

    int a1 = id1[e], a2 = id2[e], a3 = id3[e], a4 = id4[e];
    v4f v1 = __builtin_nontemporal_load(&m1[(long)e * 64 + q]);
    v4f v2 = __builtin_nontemporal_load(&m2[(long)e * 64 + q]);

    float* p;
    p = X + (long)a1 * D_X + c;
    atomicAdd(p + 0, v1.x); atomicAdd(p + 1, v1.y);
    atomicAdd(p + 2, v1.z); atomicAdd(p + 3, v1.w);
    p = X + (long)a3 * D_X + c;
    atomicAdd(p + 0, -v1.x); atomicAdd(p + 1, -v1.y);
    atomicAdd(p + 2, -v1.z); atomicAdd(p + 3, -v1.w);
    p = X + (long)a2 * D_X + D_EDGE + c;
    atomicAdd(p + 0, v2.x); atomicAdd(p + 1, v2.y);
    atomicAdd(p + 2, v2.z); atomicAdd(p + 3, v2.w);
    p = X + (long)a4 * D_X + D_EDGE + c;
    atomicAdd(p + 0, -v2.x); atomicAdd(p + 1, -v2.y);
    atomicAdd(p + 2, -v2.z); atomicAdd(p + 3, -v2.w);
}

// ---------------- GEMM: out[100000x256] = [X | h] @ W, fp32 WMMA 16x16x4 ----------------
// Block = 8 wave32; each wave owns ONE 16-row M tile and computes the FULL
// 256-wide output stripe (16 N-tiles, 128 accumulator VGPRs). A ([X|h]) is
// therefore read exactly once from HBM. W is staged chunk-by-chunk:
// 12 chunks of 64 K-rows x 256 cols = 64KB LDS, K-pair swizzled so each
// B frag is a single aligned ds_load_b64. Per k-step: 1 global b64 A load
// feeds 16 v_wmma.
__global__ __launch_bounds__(256)
void gemm_wmma_kernel(const float* __restrict__ X,
                      const float* __restrict__ h,
                      const float* __restrict__ W,
                      float* __restrict__ out) {
    __shared__ float Bs[KC * D_ATOM]; // 64 KB, element (k',n) at (k'>>1)*512 + n*2 + (k'&1)

    const int tid  = threadIdx.x;
    const int wave = tid >> 5;
    const int lane = tid & 31;

    const int tm    = blockIdx.x * 8 + wave;
    const bool valid = (tm < M_TILES);
    const int tmc   = valid ? tm : 0;

    // A-frag layout (f32 16x4): lane<16 -> row=lane, K={0,1}; lane>=16 -> row=lane-16, K={2,3}
    const int row   = tmc * 16 + (lane & 15);
    const int khalf = (lane >> 4) * 2;   // 0 or 2
    const int kh2   = khalf >> 1;        // 0 or 1 (K-pair index within a k-step)
    const float* Arow = X + (long)row * D_X   + khalf;
    const float* Hrow = h + (long)row * D_ATOM + khalf;

    v8f c[16];
    #pragma unroll
    for (int nt = 0; nt < 16; ++nt) c[nt] = (v8f){0.f,0.f,0.f,0.f,0.f,0.f,0.f,0.f};

    v2f* BsV = (v2f*)Bs;

    for (int chunk = 0; chunk < K_TOT / KC; ++chunk) {   // 12 chunks
        // ---- stage W[chunk*64 .. +64)[0..256) into LDS, K-pair swizzled ----
        __syncthreads();
        for (int p = tid; p < (KC / 2) * D_ATOM; p += 256) { // 8192 v2f
            int k2 = p >> 8;            // K-pair 0..31
            int n  = p & 255;
            int kg = chunk * KC + k2 * 2;
            v2f w;
            w.x = W[(long)kg * D_ATOM + n];
            w.y = W[(long)(kg + 1) * D_ATOM + n];
            BsV[k2 * D_ATOM + n] = w;   // contiguous b64 stores, conflict-free
        }
        __syncthreads();

        // ---- 16 k-steps of K=4; each: 1 A load + 16 wmma ----
        const bool fromX = (chunk < D_X / KC);           // chunks 0..7 read X
        const float* Abase = fromX ? (Arow + chunk * KC)
                                   : (Hrow + (chunk - D_X / KC) * KC);
        for (int k = 0; k < 16; ++k) {
            v2f a = __builtin_nontemporal_load((const v2f*)(Abase + k * 4));
            const v2f* Bk = BsV + (k * 2 + kh2) * D_ATOM + (lane & 15);
            #pragma unroll
            for (int nt = 0; nt < 16; ++nt) {
                v2f b = Bk[nt * 16];    // single aligned ds_load_b64
                c[nt] = __builtin_amdgcn_wmma_f32_16x16x4_f32(
                            false, a, false, b, (short)0, c[nt], false, false);
            }
        }
    }

    // ---- store: C/D layout: VGPR r: lanes 0-15 -> M=r, N=lane; lanes 16-31 -> M=r+8 ----
    if (valid) {
        const int mo = (lane >> 4) * 8;
        float* obase = out + (long)(tm * 16) * D_ATOM + (lane & 15);
        #pragma unroll
        for (int nt = 0; nt < 16; ++nt) {
            #pragma unroll
            for (int r = 0; r < 8; ++r) {
                __builtin_nontemporal_store(c[nt][r],
                    obase + (long)(r + mo) * D_ATOM + nt * 16);
            }
        }
    }
}

extern "C" void kernel_launch(void* const* d_in, const int* in_sizes, int n_in,
                              void* d_out, int out_size, void* d_ws, size_t ws_size,
                              hipStream_t stream) {
    const float* h  = (const float*)d_in[0];
    const float* m1 = (const float*)d_in[1];
    const float* m2 = (const float*)d_in[2];
    const int* id1  = (const int*)d_in[3];
    const int* id2  = (const int*)d_in[4];
    const int* id3  = (const int*)d_in[5];
    const int* id4  = (const int*)d_in[6];
    const float* W  = (const float*)d_in[7];

    float* X   = (float*)d_ws;               // N_ATOMS x 512 accumulator (204.8 MB)
    float* out = (float*)d_out;

    // 1) zero accumulator
    {
        long n4 = (long)N_ATOMS * D_X / 4;   // 12.8M float4
        int blocks = (int)((n4 + 255) / 256);
        zero_ws_kernel<<<blocks, 256, 0, stream>>>((v4f*)X, n4);
    }
    // 2) edge scatter with fp32 atomics (NT edge streams, RT accumulator)
    {
        long total = (long)N_EDGES * 64;     // 51.2M threads
        int blocks = (int)((total + 255) / 256);
        edge_scatter_kernel<<<blocks, 256, 0, stream>>>(
            (const v4f*)m1, (const v4f*)m2, id1, id2, id3, id4, X);
    }
    // 3) GEMM [X | h] @ W via f32 WMMA, full-N stripe per wave (A read once)
    {
        int blocks = (M_TILES + 7) / 8;      // 782
        gemm_wmma_kernel<<<blocks, 256, 0, stream>>>(X, h, W, out);
    }
}